// MyModel_38379827757401
// MI455X (gfx1250) — compile-verified
//
#include <hip/hip_runtime.h>
#include <hip/hip_bf16.h>
#include <math.h>

// ---------------------------------------------------------------------------
// Problem constants (match reference)
// ---------------------------------------------------------------------------
#define NXK   2048   // NX
#define MB    512    // BATCH
#define NL    20
#define EPSC  1e-9f

// GEMM tiling
#define BM    64
#define BN    128
#define BK    32
#define XSTR  36     // padded LDS row stride (floats): 144B -> 16B aligned, bank-spread
#define WSTR  36

typedef float v2f __attribute__((ext_vector_type(2)));
typedef float v8f __attribute__((ext_vector_type(8)));

// ---------------------------------------------------------------------------
// CDNA5 async global->LDS copy (ASYNCcnt-tracked DMA, no staging VGPRs).
// VDST operand = 32-bit LDS byte address; generic shared ptr low 32 bits are
// the LDS offset (flat-aperture layout).
// ---------------------------------------------------------------------------
__device__ __forceinline__ void async_copy16(const float* __restrict__ gsrc,
                                             float* lds_dst) {
    const unsigned l = (unsigned)(uintptr_t)lds_dst;
    asm volatile("global_load_async_to_lds_b128 %0, %1, off"
                 :: "v"(l), "v"(gsrc)
                 : "memory");
}

__device__ __forceinline__ void wait_async0() {
#if __has_builtin(__builtin_amdgcn_s_wait_asynccnt)
    __builtin_amdgcn_s_wait_asynccnt(0);
#else
    asm volatile("s_wait_asynccnt 0x0" ::: "memory");
#endif
}

__device__ __forceinline__ float softplusf(float x) {
    return log1pf(expf(x));   // jax.nn.softplus
}

// Prox of u -> -eta*(log u + log(1-u)) at x; faithful Cardano port.
__device__ __forceinline__ float prox_cardan(float eta, float x) {
    const float b = -(1.0f + x);
    const float c = x - 2.0f * eta;
    const float d = eta;
    const float p = c - b * b * (1.0f / 3.0f);
    const float q = 2.0f * b * b * b * (1.0f / 27.0f) - b * c * (1.0f / 3.0f) + d;
    const float disc = -4.0f * p * p * p - 27.0f * q * q;

    const float pm  = fminf(p, -1e-12f);
    const float m   = 2.0f * sqrtf(-pm * (1.0f / 3.0f));
    float arg = 3.0f * q / (pm * m);
    arg = fminf(fmaxf(arg, -1.0f), 1.0f);
    const float th = acosf(arg);
    const float TWO_PI = 6.28318530717958647692f;

    float root3 = 0.0f, r0 = 0.0f;
    bool found = false;
#pragma unroll
    for (int k = 0; k < 3; ++k) {
        const float t = m * cosf((th - TWO_PI * (float)k) * (1.0f / 3.0f));
        const float r = t - b * (1.0f / 3.0f);
        if (k == 0) r0 = r;
        const bool in01 = (r > 0.0f) && (r < 1.0f);
        if (in01 && !found) { root3 = r; found = true; }
    }
    if (!found) root3 = r0;   // argmax of all-false -> index 0

    const float s = sqrtf(fmaxf(q * q * 0.25f + p * p * p * (1.0f / 27.0f), 0.0f));
    const float root1 = cbrtf(-q * 0.5f + s) + cbrtf(-q * 0.5f - s) - b * (1.0f / 3.0f);

    const float u = (disc >= 0.0f) ? root3 : root1;
    return fminf(fmaxf(u, EPSC), 1.0f - EPSC);
}

// ---------------------------------------------------------------------------
// Fused GEMM step.  Out[b,i] = sum_j W[i,j] * Xin[b,j]  (+ epilogue)
//  MODE 0: W = I - g*tTT - g*reg*tDD (combined during staging);  += g*x_b
//  MODE 1: W = Pelt; epilogue = prox_cardan(g*mu_l, .)
//  MODE 2: W = Peig; plain
// V_WMMA_F32_16X16X4_F32; 8 wave32s/WG; 64x128 WG tile; 32x32 wave tile;
// double-buffered LDS fed by async global->LDS DMA.
// ---------------------------------------------------------------------------
template <int MODE>
__global__ __launch_bounds__(256)
void gemm_step(const float* __restrict__ Xin,
               const float* __restrict__ M1,     // tTT | Pelt | Peig
               const float* __restrict__ M2,     // tDD | unused
               const float* __restrict__ xb,     // x_b | unused
               const float* __restrict__ gamma_p,
               const float* __restrict__ mu_p,
               const float* __restrict__ reg_p,
               int layer,
               float* __restrict__ Out) {
    __shared__ float Xs[2][BM * XSTR];
    __shared__ float Ws[2][BN * WSTR];

    const int tid   = threadIdx.x;
    const int lane  = tid & 31;
    const int wave  = tid >> 5;
    const int waveM = wave & 1;     // 2 M-groups of 32
    const int waveN = wave >> 1;    // 4 N-groups of 32

    const int n0 = blockIdx.x * BN; // output-column block (i index)
    const int m0 = blockIdx.y * BM; // batch-row block

    // per-layer scalars
    float g = 0.0f, regl = 0.0f, eta = 0.0f;
    if (MODE == 0) {
        g    = softplusf(gamma_p[layer]);
        regl = softplusf(reg_p[layer]) * (1.0f / (10.0f * 10.0f)); // /EIGMAX^2
    } else if (MODE == 1) {
        const float gg = softplusf(gamma_p[layer]);
        eta = gg * softplusf(mu_p[layer]) * 1e-6f;
    }

    // per-thread staging coordinates (8 float4 per 32-wide row)
    const int srX[2] = { (tid + 0)   >> 3, (tid + 256) >> 3 };
    const int scX    = (tid & 7) << 2;

    // X tile: 2 async b128 per thread
    auto issueX = [&](int kk, int buf) {
#pragma unroll
        for (int it = 0; it < 2; ++it) {
            const int r = srX[it];
            async_copy16(&Xin[(size_t)(m0 + r) * NXK + kk + scX],
                         &Xs[buf][r * XSTR + scX]);
        }
    };
    // W tile (pure copy modes): 4 async b128 per thread
    auto issueW = [&](int kk, int buf) {
#pragma unroll
        for (int it = 0; it < 4; ++it) {
            const int f = tid + it * 256;
            const int r = f >> 3;
            const int c4 = (f & 7) << 2;
            async_copy16(&M1[(size_t)(n0 + r) * NXK + kk + c4],
                         &Ws[buf][r * WSTR + c4]);
        }
    };

    // MODE 0: W tile combined in registers (issue loads early, commit after
    // compute so L2 latency hides behind the WMMA block).
    float4 tt[4], dd[4];
    auto loadW0 = [&](int kk) {
#pragma unroll
        for (int it = 0; it < 4; ++it) {
            const int f = tid + it * 256;
            const int r = f >> 3;
            const int c4 = (f & 7) << 2;
            tt[it] = *(const float4*)&M1[(size_t)(n0 + r) * NXK + kk + c4];
            dd[it] = *(const float4*)&M2[(size_t)(n0 + r) * NXK + kk + c4];
        }
    };
    auto commitW0 = [&](int kk, int buf) {
#pragma unroll
        for (int it = 0; it < 4; ++it) {
            const int f = tid + it * 256;
            const int r = f >> 3;
            const int c4 = (f & 7) << 2;
            float4 w;
            w.x = -g * (tt[it].x + regl * dd[it].x);
            w.y = -g * (tt[it].y + regl * dd[it].y);
            w.z = -g * (tt[it].z + regl * dd[it].z);
            w.w = -g * (tt[it].w + regl * dd[it].w);
            const int dj = (n0 + r) - (kk + c4);   // identity on the diagonal
            if (dj >= 0 && dj < 4) ((float*)&w)[dj] += 1.0f;
            *(float4*)&Ws[buf][r * WSTR + c4] = w;
        }
    };

    v8f acc00 = {}, acc01 = {}, acc10 = {}, acc11 = {};

    // ---- prologue: stage tile 0 into buffer 0 ----
    issueX(0, 0);
    if constexpr (MODE == 0) { loadW0(0); commitW0(0, 0); }
    else                     { issueW(0, 0); }
    wait_async0();
    __syncthreads();

    const int mrow  = waveM * 32 + (lane & 15);
    const int nrow  = waveN * 32 + (lane & 15);
    const int khalf = (lane >> 4) * 2;    // f32 frag: VGPR pair = adjacent K

    for (int k0 = 0; k0 < NXK; k0 += BK) {
        const int cb = (k0 >> 5) & 1;
        const int nb = cb ^ 1;
        const bool more = (k0 + BK) < NXK;

        // ---- issue next tile (async DMA / register loads) ----
        if (more) {
            issueX(k0 + BK, nb);
            if constexpr (MODE == 0) loadW0(k0 + BK);
            else                     issueW(k0 + BK, nb);
        }

        // ---- 32x32 wave tile over current K-slab: 32 WMMAs ----
        const float* xsb = &Xs[cb][0];
        const float* wsb = &Ws[cb][0];
#pragma unroll
        for (int km = 0; km < BK / 4; ++km) {
            const int kcol = km * 4 + khalf;
            const v2f a0 = *(const v2f*)&xsb[(mrow)      * XSTR + kcol];
            const v2f a1 = *(const v2f*)&xsb[(mrow + 16) * XSTR + kcol];
            const v2f b0 = *(const v2f*)&wsb[(nrow)      * WSTR + kcol];
            const v2f b1 = *(const v2f*)&wsb[(nrow + 16) * WSTR + kcol];
            acc00 = __builtin_amdgcn_wmma_f32_16x16x4_f32(false, a0, false, b0,
                                                          (short)0, acc00, false, false);
            acc01 = __builtin_amdgcn_wmma_f32_16x16x4_f32(false, a0, false, b1,
                                                          (short)0, acc01, false, false);
            acc10 = __builtin_amdgcn_wmma_f32_16x16x4_f32(false, a1, false, b0,
                                                          (short)0, acc10, false, false);
            acc11 = __builtin_amdgcn_wmma_f32_16x16x4_f32(false, a1, false, b1,
                                                          (short)0, acc11, false, false);
        }

        // ---- commit combined W tile after compute (MODE 0 only) ----
        if (more) {
            if constexpr (MODE == 0) commitW0(k0 + BK, nb);
        }
        wait_async0();
        __syncthreads();
    }

    // ---- epilogue: C/D layout -> VGPR v, lanes0-15: M=v, lanes16-31: M=v+8
    const int lrow = (lane >> 4) * 8;
    const int lcol = lane & 15;
    const int mbase = m0 + waveM * 32;
    const int nbase = n0 + waveN * 32;
#pragma unroll
    for (int mt = 0; mt < 2; ++mt) {
#pragma unroll
        for (int nt = 0; nt < 2; ++nt) {
            const v8f a = (mt == 0) ? ((nt == 0) ? acc00 : acc01)
                                    : ((nt == 0) ? acc10 : acc11);
#pragma unroll
            for (int v = 0; v < 8; ++v) {
                const int mm = mbase + mt * 16 + lrow + v;
                const int nn = nbase + nt * 16 + lcol;
                float val = a[v];
                if (MODE == 0) val += g * xb[(size_t)mm * NXK + nn];
                if (MODE == 1) val = prox_cardan(eta, val);
                Out[(size_t)mm * NXK + nn] = val;
            }
        }
    }
}

// ---------------------------------------------------------------------------
// Driver: 20 layers x (GEMM1-combined, GEMM2+prox, GEMM3), ping-pong ws bufs.
// ---------------------------------------------------------------------------
extern "C" void kernel_launch(void* const* d_in, const int* in_sizes, int n_in,
                              void* d_out, int out_size, void* d_ws, size_t ws_size,
                              hipStream_t stream) {
    const float* x     = (const float*)d_in[0];
    const float* x_b   = (const float*)d_in[1];
    const float* tDD   = (const float*)d_in[2];
    const float* tTT   = (const float*)d_in[3];
    const float* Peig  = (const float*)d_in[4];
    const float* Pelt  = (const float*)d_in[5];
    const float* gamma = (const float*)d_in[6];
    const float* mu    = (const float*)d_in[7];
    const float* reg   = (const float*)d_in[8];
    float* out = (float*)d_out;

    const size_t nelem = (size_t)MB * NXK;
    float* bufA = (float*)d_ws;
    float* bufB = bufA + nelem;

    dim3 grid(NXK / BN, MB / BM);   // 16 x 8 = 128 workgroups
    dim3 block(256);

    const float* cur = x;
    for (int l = 0; l < NL; ++l) {
        float* y   = bufA;
        float* z   = bufB;
        float* nxt = (l == NL - 1) ? out : bufA;

        gemm_step<0><<<grid, block, 0, stream>>>(cur, tTT, tDD, x_b,
                                                 gamma, mu, reg, l, y);
        gemm_step<1><<<grid, block, 0, stream>>>(y, Pelt, nullptr, nullptr,
                                                 gamma, mu, reg, l, z);
        gemm_step<2><<<grid, block, 0, stream>>>(z, Peig, nullptr, nullptr,
                                                 gamma, mu, reg, l, nxt);
        cur = nxt;
        float* tmp = bufA; bufA = bufB; bufB = tmp;
    }
}